// NoiseLayer_48352741818664
// MI455X (gfx1250) — compile-verified
//
#include <hip/hip_runtime.h>
#include <stdint.h>

// Streaming bit-flip noise + clip for MI455X (gfx1250).
// HBM-bound: 268 MB traffic, ~11.5 us floor @ 23.3 TB/s.
// Data path: global -> LDS via global_load_async_to_lds_b128 (NT, double
// buffered, 2 x b128 per thread per stage, ASYNCcnt pipelining),
// LDS -> VGPR (ds_load_b128), VALU RNG+clip, NT b128 stores.

typedef float f4 __attribute__((ext_vector_type(4)));

#define NOISE_SEED 0x2545F4914F6CDD1Dull
#define BLOCK_THREADS 256
#define GROUPS_PER_TILE (2 * BLOCK_THREADS)   // 512 f4 groups = 8 KB per tile

__device__ __forceinline__ uint64_t splitmix64(uint64_t z) {
  z += 0x9E3779B97F4A7C15ull;
  z = (z ^ (z >> 30)) * 0xBF58476D1CE4E5B9ull;
  z = (z ^ (z >> 27)) * 0x94D049BB133111EBull;
  return z ^ (z >> 31);
}

// Per-element 32-bit flip mask, each bit ~ Bernoulli(0.01), via
// Binomial(32,0.01) count (17-bit CDF thresholds on r[31:15]) and up to
// three flip positions from independent 5-bit fields r[4:0],r[9:5],r[14:10].
__device__ __forceinline__ uint32_t flip_mask(uint32_t r) {
  uint32_t u = r >> 15;                                   // 17-bit uniform
  uint32_t m = 0u;
  m |= (u >= 95027u)  ? (1u << (r & 31u))         : 0u;   // P(X>=1)=0.27500
  m |= (u >= 125742u) ? (1u << ((r >> 5) & 31u))  : 0u;   // P(X>=2)=0.040655
  m |= (u >= 130552u) ? (1u << ((r >> 10) & 31u)) : 0u;   // P(X>=3)=0.0039646
  return m;
}

__device__ __forceinline__ float flip_clip(float x, uint32_t r) {
  uint32_t b = __float_as_uint(x) ^ flip_mask(r);
  float f = __uint_as_float(b);
  return fminf(fmaxf(f, -3.0f), 3.0f);
}

// Process one f4 group; RNG keyed on the global group index -> deterministic.
__device__ __forceinline__ f4 process_group(f4 v, int g) {
  const uint64_t base = ((uint64_t)(uint32_t)g) << 1;
  const uint64_t z0 = splitmix64(base ^ NOISE_SEED);
  const uint64_t z1 = splitmix64((base + 1ull) ^ NOISE_SEED);
  f4 y;
  y.x = flip_clip(v.x, (uint32_t)z0);
  y.y = flip_clip(v.y, (uint32_t)(z0 >> 32));
  y.z = flip_clip(v.z, (uint32_t)z1);
  y.w = flip_clip(v.w, (uint32_t)(z1 >> 32));
  return y;
}

__global__ __launch_bounds__(BLOCK_THREADS)
void noise_bitflip_clip_kernel(const float* __restrict__ x,
                               float* __restrict__ out,
                               int ntiles, int n4, int n) {
  __shared__ f4 stage[2][2][BLOCK_THREADS];   // 16 KB: [buf][half][thread]

  const int tid = threadIdx.x;
  // Generic LDS pointer low 32 bits == wave-relative LDS byte address.
  const uint32_t a0 = (uint32_t)(uintptr_t)&stage[0][0][tid];
  const uint32_t b0 = (uint32_t)(uintptr_t)&stage[0][1][tid];
  const uint32_t a1 = (uint32_t)(uintptr_t)&stage[1][0][tid];
  const uint32_t b1 = (uint32_t)(uintptr_t)&stage[1][1][tid];

  int t = blockIdx.x;

  // Prologue: fill stage 0 (two NT async b128 per thread; ASYNCcnt=2).
  if (t < ntiles) {
    uint32_t off = ((uint32_t)t * GROUPS_PER_TILE + (uint32_t)tid) * 16u;
    asm volatile("global_load_async_to_lds_b128 %0, %1, %2 th:TH_LOAD_NT"
                 :: "v"(a0), "v"(off), "s"(x) : "memory");
    asm volatile("global_load_async_to_lds_b128 %0, %1, %2 th:TH_LOAD_NT"
                 :: "v"(b0), "v"(off + BLOCK_THREADS * 16u), "s"(x) : "memory");
  }

  int buf = 0;
  for (; t < ntiles; t += gridDim.x) {      // block-uniform hot loop
    const int nt = t + gridDim.x;
    if (nt < ntiles) {
      // Prefetch next tile into the other buffer, wait only for this tile.
      uint32_t off = ((uint32_t)nt * GROUPS_PER_TILE + (uint32_t)tid) * 16u;
      asm volatile("global_load_async_to_lds_b128 %0, %1, %2 th:TH_LOAD_NT"
                   :: "v"(buf ? a0 : a1), "v"(off), "s"(x) : "memory");
      asm volatile("global_load_async_to_lds_b128 %0, %1, %2 th:TH_LOAD_NT"
                   :: "v"(buf ? b0 : b1), "v"(off + BLOCK_THREADS * 16u), "s"(x)
                   : "memory");
      asm volatile("s_wait_asynccnt 0x2" ::: "memory");
    } else {
      asm volatile("s_wait_asynccnt 0x0" ::: "memory");
    }

    const int gA = t * GROUPS_PER_TILE + tid;
    const int gB = gA + BLOCK_THREADS;
    f4 vA = stage[buf][0][tid];             // ds_load_b128 (own slot, no barrier)
    f4 vB = stage[buf][1][tid];

    __builtin_nontemporal_store(process_group(vA, gA), (f4*)out + gA);
    __builtin_nontemporal_store(process_group(vB, gB), (f4*)out + gB);
    buf ^= 1;
  }

  // Remainder path (dead for the reference shape n = 2^25): block 0 handles
  // the partial tile with plain vector loads, then thread 0 the n%4 scalars.
  if (blockIdx.x == 0) {
    for (int g = ntiles * GROUPS_PER_TILE + tid; g < n4; g += BLOCK_THREADS) {
      f4 v = __builtin_nontemporal_load((const f4*)x + g);
      __builtin_nontemporal_store(process_group(v, g), (f4*)out + g);
    }
    if (tid == 0) {
      for (int e = n4 * 4; e < n; ++e) {
        uint64_t z = splitmix64(((uint64_t)(uint32_t)e * 0x9E3779B1ull) ^ NOISE_SEED);
        out[e] = flip_clip(x[e], (uint32_t)z);
      }
    }
  }
}

extern "C" void kernel_launch(void* const* d_in, const int* in_sizes, int n_in,
                              void* d_out, int out_size, void* d_ws, size_t ws_size,
                              hipStream_t stream) {
  (void)n_in; (void)d_ws; (void)ws_size; (void)out_size;
  const float* x = (const float*)d_in[0];
  float* out = (float*)d_out;
  const int n = in_sizes[0];
  const int n4 = n >> 2;
  const int ntiles = n4 / GROUPS_PER_TILE;  // full 8 KB tiles (16384 for ref)

  int blocks = ntiles;
  if (blocks > 4096) blocks = 4096;         // grid-stride: 4 tiles/block for ref
  if (blocks < 1) blocks = 1;

  noise_bitflip_clip_kernel<<<blocks, BLOCK_THREADS, 0, stream>>>(
      x, out, ntiles, n4, n);
}